// LipResConv_13013750907152
// MI455X (gfx1250) — compile-verified
//
#include <hip/hip_runtime.h>
#include <math.h>

// ---------------------------------------------------------------------------
// LipResConv (Cayley orthogonal FFT conv) for MI455X / gfx1250.
// Heavy math = 64x64(x32) complex GEMMs on V_WMMA_F32_16X16X4_F32.
// Global->LDS staging via GLOBAL_LOAD_ASYNC_TO_LDS_* (ASYNCcnt).
// Wave32, 256-thread workgroups (8 waves), LDS-resident tiles.
// ---------------------------------------------------------------------------

#define NB 256
#define LDT 66    // padded stride (float2 units) for 64-wide tiles (66*8 = 16B-aligned rows)
#define LD32 34   // padded stride for 32-wide tiles  (34*8 = 16B-aligned rows)

typedef __attribute__((ext_vector_type(2))) float v2f;
typedef __attribute__((ext_vector_type(8))) float v8f;

__device__ __forceinline__ float2 cmul(float2 a, float2 b) {
  return make_float2(a.x * b.x - a.y * b.y, a.x * b.y + a.y * b.x);
}
__device__ __forceinline__ float2 csub(float2 a, float2 b) {
  return make_float2(a.x - b.x, a.y - b.y);
}
__device__ __forceinline__ float2 cconj(float2 a) { return make_float2(a.x, -a.y); }

__device__ __forceinline__ v8f wmma4(v2f a, v2f b, v8f c) {
  // 8 args: (neg_a, A, neg_b, B, c_mod, C, reuse_a, reuse_b)
  return __builtin_amdgcn_wmma_f32_16x16x4_f32(false, a, false, b, (short)0, c,
                                               false, false);
}

// ---- async global->LDS helpers (CDNA5 ASYNCcnt path) -----------------------
__device__ __forceinline__ unsigned lds_off(const void* p) {
  // addrspace(3) offsets occupy the low 32 bits of the generic pointer
  return (unsigned)(uintptr_t)p;
}
__device__ __forceinline__ void async_g2s_b128(void* lds, const void* g) {
  asm volatile("global_load_async_to_lds_b128 %0, %1, off"
               :
               : "v"(lds_off(lds)), "v"((unsigned long long)(uintptr_t)g)
               : "memory");
}
__device__ __forceinline__ void async_g2s_b64(void* lds, const void* g) {
  asm volatile("global_load_async_to_lds_b64 %0, %1, off"
               :
               : "v"(lds_off(lds)), "v"((unsigned long long)(uintptr_t)g)
               : "memory");
}
__device__ __forceinline__ void wait_async0() {
  asm volatile("s_wait_asynccnt 0" ::: "memory");
}

// D = scale * (A x B), complex, all operands in LDS (float2, padded strides).
// M, Ncols multiples of 16; Kdim multiple of 4. 8 waves cooperate; caller
// must __syncthreads() before (operands ready) and after (D complete).
__device__ __forceinline__ void wg_cgemm(const float2* __restrict__ A, int lda,
                                         const float2* __restrict__ B, int ldb,
                                         float2* __restrict__ D, int ldd,
                                         int M, int Kdim, int Ncols, float scale,
                                         int wave, int lane) {
  const int tilesN = Ncols >> 4;
  const int ntiles = (M >> 4) * tilesN;
  const int r = lane & 15;          // A row-in-tile / B,D col-in-tile
  const int kb = (lane >> 4) << 1;  // K sub-offset {0,2}
  for (int t = wave; t < ntiles; t += 8) {
    const int tm = (t / tilesN) << 4;
    const int tn = (t % tilesN) << 4;
    v8f cr = {};
    v8f ci = {};
    for (int k = 0; k < Kdim; k += 4) {
      const float2 a0 = A[(tm + r) * lda + (k + kb)];
      const float2 a1 = A[(tm + r) * lda + (k + kb + 1)];
      const float2 b0 = B[(k + kb) * ldb + (tn + r)];
      const float2 b1 = B[(k + kb + 1) * ldb + (tn + r)];
      v2f ar = {a0.x, a1.x};
      v2f ai = {a0.y, a1.y};
      v2f br = {b0.x, b1.x};
      v2f bi = {b0.y, b1.y};
      v2f nbi = {-b0.y, -b1.y};  // f32 WMMA has no A/B negate; do it in VALU
      cr = wmma4(ar, br, cr);
      cr = wmma4(ai, nbi, cr);
      ci = wmma4(ar, bi, ci);
      ci = wmma4(ai, br, ci);
    }
    const int n = tn + (lane & 15);
    const int mbase = tm + ((lane >> 4) << 3);
#pragma unroll
    for (int v = 0; v < 8; ++v) {
      D[(mbase + v) * ldd + n] = make_float2(cr[v] * scale, ci[v] * scale);
    }
  }
}

// Gauss-Jordan inverse of 64x64 complex M (stride LDT) -> Inv. Destroys M.
// I+A with Hermitian part >= I: no pivoting needed. fac: 64 float2 scratch.
__device__ __forceinline__ void wg_cinv64(float2* __restrict__ Mm,
                                          float2* __restrict__ Inv,
                                          float2* __restrict__ fac, int tid) {
  for (int i = tid; i < 64 * 64; i += NB) {
    int r = i >> 6, c = i & 63;
    Inv[r * LDT + c] = make_float2(r == c ? 1.f : 0.f, 0.f);
  }
  __syncthreads();
  for (int p = 0; p < 64; ++p) {
    float2 piv = Mm[p * LDT + p];
    float id = 1.f / (piv.x * piv.x + piv.y * piv.y);
    float2 ip = make_float2(piv.x * id, -piv.y * id);
    __syncthreads();
    if (tid < 128) {  // scale pivot row of [M | Inv]
      int c = tid & 63;
      float2* row = (tid < 64) ? (Mm + p * LDT) : (Inv + p * LDT);
      row[c] = cmul(row[c], ip);
    }
    __syncthreads();
    if (tid < 64) fac[tid] = Mm[tid * LDT + p];
    __syncthreads();
    for (int i = tid; i < 64 * 128; i += NB) {
      int r = i >> 7, c = i & 127;
      if (r == p) continue;
      float2 f = fac[r];
      if (c < 64) {
        Mm[r * LDT + c] = csub(Mm[r * LDT + c], cmul(f, Mm[p * LDT + c]));
      } else {
        int cc = c - 64;
        Inv[r * LDT + cc] = csub(Inv[r * LDT + cc], cmul(f, Inv[p * LDT + cc]));
      }
    }
    __syncthreads();
  }
}

// --------------------------- kernels ---------------------------------------

__global__ void build_dft_kernel(float2* __restrict__ F, float2* __restrict__ Fi) {
  const float TWO_PI = 6.28318530717958647692f;
  for (int i = threadIdx.x; i < 4096; i += NB) {
    int u = i >> 6, k = i & 63;
    float s, c;
    __sincosf(-TWO_PI * (float)(u * k) / 64.f, &s, &c);
    F[i] = make_float2(c, s);                      // e^{-2pi i uk/64}
    Fi[i] = make_float2(c / 64.f, -s / 64.f);      // conj(F)/64
  }
}

// One WG per (b,c) image: Y = F * X * F (F symmetric). dst[f][c][b].
__global__ void fft2_kernel(const float* __restrict__ src,
                            const float2* __restrict__ Ftab,
                            float2* __restrict__ dst) {
  extern __shared__ __align__(16) float2 smem[];
  float2* Fm = smem;
  float2* X = Fm + 64 * LDT;
  float2* T = X + 64 * LDT;
  const int tid = threadIdx.x, wave = tid >> 5, lane = tid & 31;
  const int img = blockIdx.x;  // b*64 + c
  const int b = img >> 6, c = img & 63;
  const float* sp = src + (size_t)img * 4096;
  // F table: async B128 (rows are 16B-aligned chunks of 2 float2)
  for (int ch = tid; ch < 2048; ch += NB) {
    int r = ch >> 5, cp = (ch & 31) << 1;
    async_g2s_b128(&Fm[r * LDT + cp], &Ftab[r * 64 + cp]);
  }
  for (int i = tid; i < 4096; i += NB) {  // real -> complex, needs VALU
    X[(i >> 6) * LDT + (i & 63)] = make_float2(sp[i], 0.f);
  }
  wait_async0();
  __syncthreads();
  wg_cgemm(Fm, LDT, X, LDT, T, LDT, 64, 64, 64, 1.f, wave, lane);
  __syncthreads();
  wg_cgemm(T, LDT, Fm, LDT, X, LDT, 64, 64, 64, 1.f, wave, lane);
  __syncthreads();
  for (int i = tid; i < 4096; i += NB) {
    dst[(size_t)i * 2048 + c * 32 + b] = X[(i >> 6) * LDT + (i & 63)];
  }
}

// One WG per frequency: Cayley transform + fused per-frequency products.
__global__ void cayley_kernel(const float* __restrict__ W,
                              const float2* __restrict__ xfft,
                              float2* __restrict__ GTsHg,
                              float2* __restrict__ Yf1,   // GTHx freq
                              float2* __restrict__ Yf2) { // Hx freq
  extern __shared__ __align__(16) float2 smem[];
  float2* U = smem;              // GTs later
  float2* V = U + 64 * LDT;      // x tile later
  float2* B2 = V + 64 * LDT;     // VH / I+A / I-A / HTsH
  float2* B3 = B2 + 64 * LDT;    // A / HTs / y1
  float2* B4 = B3 + 64 * LDT;    // Inv / t
  float2* fv = B4 + 64 * LDT;    // 64 pivot factors
  const int tid = threadIdx.x, wave = tid >> 5, lane = tid & 31;
  const int f = blockIdx.x;
  const int fu = f >> 6, fvv = f & 63;
  const float TWO_PI = 6.28318530717958647692f;
  float2 twu[3], twv[3];
#pragma unroll
  for (int i = 0; i < 3; ++i) {
    float s, c;
    __sincosf(TWO_PI * (float)(fu * i) / 64.f, &s, &c);
    twu[i] = make_float2(c, s);
    __sincosf(TWO_PI * (float)(fvv * i) / 64.f, &s, &c);
    twv[i] = make_float2(c, s);
  }
  float ss, sc;
  __sincosf(-TWO_PI * (float)(fu + fvv) / 64.f, &ss, &sc);
  const float2 shift = make_float2(sc, ss);
  float2 t9[9];
#pragma unroll
  for (int i = 0; i < 3; ++i)
#pragma unroll
    for (int j = 0; j < 3; ++j) t9[i * 3 + j] = cmul(shift, cmul(twu[i], twv[j]));

  // wfft rows: U = rows 0..63, V = rows 64..127 (cols = cout)
  for (int e = tid; e < 128 * 64; e += NB) {
    int r = e >> 6, o = e & 63;
    const float* wp = W + (size_t)(r * 64 + o) * 9;
    float2 acc = make_float2(0.f, 0.f);
#pragma unroll
    for (int k = 0; k < 9; ++k) {
      float w = wp[k];
      acc.x += w * t9[k].x;
      acc.y += w * t9[k].y;
    }
    if (r < 64) U[r * LDT + o] = acc; else V[(r - 64) * LDT + o] = acc;
  }
  __syncthreads();
  for (int e = tid; e < 4096; e += NB) {  // B2 = V^H
    int r = e >> 6, c = e & 63;
    B2[r * LDT + c] = cconj(V[c * LDT + r]);
  }
  __syncthreads();
  wg_cgemm(B2, LDT, V, LDT, B3, LDT, 64, 64, 64, 1.f, wave, lane);  // B3 = V^H V
  __syncthreads();
  for (int e = tid; e < 4096; e += NB) {  // A = B3 + U - U^H ; B2 = I + A
    int r = e >> 6, c = e & 63;
    float2 a = B3[r * LDT + c];
    float2 u1 = U[r * LDT + c], u2 = U[c * LDT + r];
    a.x += u1.x - u2.x;
    a.y += u1.y + u2.y;  // - conj(u2)
    B3[r * LDT + c] = a;
    B2[r * LDT + c] = make_float2(a.x + (r == c ? 1.f : 0.f), a.y);
  }
  __syncthreads();
  wg_cinv64(B2, B4, fv, tid);  // B4 = (I+A)^{-1}
  for (int e = tid; e < 4096; e += NB) {  // B2 = I - A
    int r = e >> 6, c = e & 63;
    float2 a = B3[r * LDT + c];
    B2[r * LDT + c] = make_float2((r == c ? 1.f : 0.f) - a.x, -a.y);
  }
  __syncthreads();
  wg_cgemm(B4, LDT, B2, LDT, U, LDT, 64, 64, 64, 1.f, wave, lane);   // U = GTs
  __syncthreads();
  wg_cgemm(V, LDT, B4, LDT, B3, LDT, 64, 64, 64, -2.f, wave, lane);  // B3 = HTs
  __syncthreads();
  float2* gout = GTsHg + (size_t)f * 4096;
  for (int e = tid; e < 4096; e += NB) {  // store GTs^H ; B2 = HTs^H
    int r = e >> 6, c = e & 63;
    gout[e] = cconj(U[c * LDT + r]);
    B2[r * LDT + c] = cconj(B3[c * LDT + r]);
  }
  __syncthreads();
  float2* Xt = V;  // 64 x 32 tile, stride LD32
  const float2* xp = xfft + (size_t)f * 2048;
  for (int ch = tid; ch < 1024; ch += NB) {  // async B128 tile fill
    int r = ch >> 4, cp = (ch & 15) << 1;
    async_g2s_b128(&Xt[r * LD32 + cp], &xp[r * 32 + cp]);
  }
  wait_async0();
  __syncthreads();
  wg_cgemm(B2, LDT, Xt, LD32, B4, LD32, 64, 64, 32, 1.f, wave, lane);  // t = HTsH x
  __syncthreads();
  float2* y2 = Yf2 + (size_t)f * 2048;
  for (int e = tid; e < 2048; e += NB) y2[e] = B4[(e >> 5) * LD32 + (e & 31)];
  __syncthreads();
  wg_cgemm(U, LDT, B4, LD32, B3, LD32, 64, 64, 32, 1.f, wave, lane);   // y1 = GTs t
  __syncthreads();
  float2* y1 = Yf1 + (size_t)f * 2048;
  for (int e = tid; e < 2048; e += NB) y1[e] = B3[(e >> 5) * LD32 + (e & 31)];
}

// One WG per (b,o): Z = Fi * Y * Fi; mode=1 -> tanh(-(1/Lam)*Re(Z)+bias), else Re(Z).
__global__ void ifft2_post_kernel(const float2* __restrict__ src,
                                  const float2* __restrict__ Fitab,
                                  float* __restrict__ dst,
                                  const float* __restrict__ lamb,
                                  const float* __restrict__ bias, int mode) {
  extern __shared__ __align__(16) float2 smem[];
  float2* Fm = smem;
  float2* Y = Fm + 64 * LDT;
  float2* T = Y + 64 * LDT;
  const int tid = threadIdx.x, wave = tid >> 5, lane = tid & 31;
  const int img = blockIdx.x;  // b*64 + o
  const int b = img >> 6, o = img & 63;
  for (int ch = tid; ch < 2048; ch += NB) {  // Fi table: async B128
    int r = ch >> 5, cp = (ch & 31) << 1;
    async_g2s_b128(&Fm[r * LDT + cp], &Fitab[r * 64 + cp]);
  }
  for (int i = tid; i < 4096; i += NB) {     // strided gather: async B64
    async_g2s_b64(&Y[(i >> 6) * LDT + (i & 63)],
                  &src[(size_t)i * 2048 + o * 32 + b]);
  }
  wait_async0();
  __syncthreads();
  wg_cgemm(Fm, LDT, Y, LDT, T, LDT, 64, 64, 64, 1.f, wave, lane);
  __syncthreads();
  wg_cgemm(T, LDT, Fm, LDT, Y, LDT, 64, 64, 64, 1.f, wave, lane);
  __syncthreads();
  float mul = 0.f, bb = 0.f;
  if (mode) {
    mul = -1.0f / (0.5f + __expf(lamb[o]));  // L_CONST = 1
    bb = bias[o];
  }
  float* dp = dst + (size_t)img * 4096;
  for (int i = tid; i < 4096; i += NB) {
    float zr = Y[(i >> 6) * LDT + (i & 63)].x;
    dp[i] = mode ? tanhf(mul * zr + bb) : zr;
  }
}

// One WG per frequency: outf[f] = GTsH[f] @ sfft[f]  (64x64 x 64x32 complex).
__global__ void freq_apply_kernel(const float2* __restrict__ GTsHg,
                                  const float2* __restrict__ sfft,
                                  float2* __restrict__ outf) {
  extern __shared__ __align__(16) float2 smem[];
  float2* G = smem;             // 64 x LDT
  float2* S = G + 64 * LDT;     // 64 x LD32
  float2* D = S + 64 * LD32;    // 64 x LD32
  const int tid = threadIdx.x, wave = tid >> 5, lane = tid & 31;
  const int f = blockIdx.x;
  const float2* gp = GTsHg + (size_t)f * 4096;
  for (int ch = tid; ch < 2048; ch += NB) {  // 32KB GTsH tile: async B128
    int r = ch >> 5, cp = (ch & 31) << 1;
    async_g2s_b128(&G[r * LDT + cp], &gp[r * 64 + cp]);
  }
  const float2* sp = sfft + (size_t)f * 2048;
  for (int ch = tid; ch < 1024; ch += NB) {  // 16KB sigfft tile: async B128
    int r = ch >> 4, cp = (ch & 15) << 1;
    async_g2s_b128(&S[r * LD32 + cp], &sp[r * 32 + cp]);
  }
  wait_async0();
  __syncthreads();
  wg_cgemm(G, LDT, S, LD32, D, LD32, 64, 64, 32, 1.f, wave, lane);
  __syncthreads();
  float2* op = outf + (size_t)f * 2048;
  for (int i = tid; i < 2048; i += NB) op[i] = D[(i >> 5) * LD32 + (i & 31)];
}

// One WG per (b,o): y = Hx + Re(ifft2(Gsig)), subsample [::2,::2].
__global__ void final_kernel(const float2* __restrict__ Gsigf,
                             const float2* __restrict__ Fitab,
                             const float* __restrict__ Hx,
                             float* __restrict__ out) {
  extern __shared__ __align__(16) float2 smem[];
  float2* Fm = smem;
  float2* Y = Fm + 64 * LDT;
  float2* T = Y + 64 * LDT;
  const int tid = threadIdx.x, wave = tid >> 5, lane = tid & 31;
  const int img = blockIdx.x;  // b*64 + o
  const int b = img >> 6, o = img & 63;
  for (int ch = tid; ch < 2048; ch += NB) {  // Fi table: async B128
    int r = ch >> 5, cp = (ch & 31) << 1;
    async_g2s_b128(&Fm[r * LDT + cp], &Fitab[r * 64 + cp]);
  }
  for (int i = tid; i < 4096; i += NB) {     // strided gather: async B64
    async_g2s_b64(&Y[(i >> 6) * LDT + (i & 63)],
                  &Gsigf[(size_t)i * 2048 + o * 32 + b]);
  }
  wait_async0();
  __syncthreads();
  wg_cgemm(Fm, LDT, Y, LDT, T, LDT, 64, 64, 64, 1.f, wave, lane);
  __syncthreads();
  wg_cgemm(T, LDT, Fm, LDT, Y, LDT, 64, 64, 64, 1.f, wave, lane);
  __syncthreads();
  const float* hp = Hx + (size_t)img * 4096;
  float* op = out + (size_t)img * 1024;
  for (int i = tid; i < 1024; i += NB) {
    int r = (i >> 5) << 1, c = (i & 31) << 1;
    op[i] = hp[r * 64 + c] + Y[r * LDT + c].x;  // L_CONST = 1
  }
}

// --------------------------- launcher --------------------------------------

extern "C" void kernel_launch(void* const* d_in, const int* in_sizes, int n_in,
                              void* d_out, int out_size, void* d_ws, size_t ws_size,
                              hipStream_t stream) {
  (void)in_sizes; (void)n_in; (void)out_size; (void)ws_size;
  const float* x = (const float*)d_in[0];     // (32,64,64,64)
  const float* W = (const float*)d_in[1];     // (128,64,3,3)
  const float* lamb = (const float*)d_in[2];  // (64,)
  const float* bias = (const float*)d_in[3];  // (64,)
  float* out = (float*)d_out;                 // (32,64,32,32)

  char* ws = (char*)d_ws;
  size_t off = 0;
  auto alloc = [&](size_t bytes) -> void* {
    void* p = ws + off;
    off += (bytes + 255) & ~(size_t)255;
    return p;
  };
  float2* Ftab  = (float2*)alloc((size_t)4096 * sizeof(float2));
  float2* Fitab = (float2*)alloc((size_t)4096 * sizeof(float2));
  float2* xfft  = (float2*)alloc((size_t)4096 * 2048 * sizeof(float2));  // reused: sigfft
  float2* GTsH  = (float2*)alloc((size_t)4096 * 4096 * sizeof(float2));
  float2* Yf1   = (float2*)alloc((size_t)4096 * 2048 * sizeof(float2));  // reused: Gsigf
  float2* Yf2   = (float2*)alloc((size_t)4096 * 2048 * sizeof(float2));
  float* sig    = (float*)alloc((size_t)2048 * 4096 * sizeof(float));
  float* Hx     = (float*)alloc((size_t)2048 * 4096 * sizeof(float));

  const size_t smFFT = (size_t)3 * 64 * LDT * sizeof(float2);
  const size_t smCay = (size_t)5 * 64 * LDT * sizeof(float2) + 64 * sizeof(float2);
  const size_t smFrq = ((size_t)64 * LDT + 2 * (size_t)64 * LD32) * sizeof(float2);

  build_dft_kernel<<<1, NB, 0, stream>>>(Ftab, Fitab);
  fft2_kernel<<<2048, NB, smFFT, stream>>>(x, Ftab, xfft);
  cayley_kernel<<<4096, NB, smCay, stream>>>(W, xfft, GTsH, Yf1, Yf2);
  ifft2_post_kernel<<<2048, NB, smFFT, stream>>>(Yf1, Fitab, sig, lamb, bias, 1);
  ifft2_post_kernel<<<2048, NB, smFFT, stream>>>(Yf2, Fitab, Hx, lamb, bias, 0);
  fft2_kernel<<<2048, NB, smFFT, stream>>>(sig, Ftab, xfft);           // sigfft
  freq_apply_kernel<<<4096, NB, smFrq, stream>>>(GTsH, xfft, Yf1);     // Gsigf
  final_kernel<<<2048, NB, smFFT, stream>>>(Yf1, Fitab, Hx, out);
}